// NeighborSearch_36112085024913
// MI455X (gfx1250) — compile-verified
//
#include <hip/hip_runtime.h>
#include <hip/hip_bf16.h>

typedef __attribute__((ext_vector_type(2))) float v2f;
typedef __attribute__((ext_vector_type(8))) float v8f;

__device__ __forceinline__ unsigned ballot32(bool p) {
#if __has_builtin(__builtin_amdgcn_ballot_w32)
  return (unsigned)__builtin_amdgcn_ballot_w32(p);
#else
  return (unsigned)__ballot(p);
#endif
}

// ---------------------------------------------------------------------------
// Kernel 1: per-row neighbor counts via V_WMMA_F32_16X16X4_F32.
// One wave (= one workgroup of 32) owns a 16-row tile and sweeps all N
// columns in 16-col tiles. A-matrix (16x4): row m = (qx, qy, qz, q2).
// B-matrix (4x16): col n = (-2dx, -2dy, -2dz, 1). C = broadcast d2 per
// column. D = squared distance tile.
// A layout: lanes 0-15 hold K=0,1 in VGPR0,1; lanes 16-31 hold K=2,3.
// C/D layout: VGPR k: lanes 0-15 -> row k, lanes 16-31 -> row k+8, N=lane%16.
// ---------------------------------------------------------------------------
__global__ __launch_bounds__(32) void ns_count_kernel(
    const float* __restrict__ data, const float* __restrict__ queries,
    const float* __restrict__ radi, int* __restrict__ counts,
    int Mq, int Nd) {
  const int lane = threadIdx.x & 31;
  const int rb = blockIdx.x * 16;
  if (rb >= Mq) return;  // wave-uniform
  const int half = lane >> 4;
  const int lr = lane & 15;

  // A operand (fixed for this wave)
  const int m = rb + lr;
  const float qx = queries[m * 3 + 0];
  const float qy = queries[m * 3 + 1];
  const float qz = queries[m * 3 + 2];
  const float q2 = qx * qx + qy * qy + qz * qz;
  v2f a;
  a.x = half ? qz : qx;
  a.y = half ? q2 : qy;

  // per-row squared radii, indexed to match the C/D accumulator layout
  float r2v[8];
#pragma unroll
  for (int k = 0; k < 8; ++k) {
    float r = radi[rb + k + 8 * half];
    r2v[k] = r * r;
  }

  unsigned cnt[8];
#pragma unroll
  for (int k = 0; k < 8; ++k) cnt[k] = 0u;

  for (int cb = 0; cb < Nd; cb += 16) {
    const int j = cb + lr;
    __builtin_prefetch(&data[(j + 64) * 3], 0, 1);
    const float dx = data[j * 3 + 0];
    const float dy = data[j * 3 + 1];
    const float dz = data[j * 3 + 2];
    const float d2 = dx * dx + dy * dy + dz * dz;
    v2f b;
    b.x = half ? (-2.0f * dz) : (-2.0f * dx);
    b.y = half ? 1.0f : (-2.0f * dy);
    v8f c;
#pragma unroll
    for (int k = 0; k < 8; ++k) c[k] = d2;
    v8f acc = __builtin_amdgcn_wmma_f32_16x16x4_f32(
        false, a, false, b, (short)0, c, false, false);
#pragma unroll
    for (int k = 0; k < 8; ++k) cnt[k] += (acc[k] <= r2v[k]) ? 1u : 0u;
  }

  // reduce across the 16 lanes of each half (xor<16 stays within a half)
#pragma unroll
  for (int k = 0; k < 8; ++k) {
#pragma unroll
    for (int off = 1; off < 16; off <<= 1)
      cnt[k] += (unsigned)__shfl_xor((int)cnt[k], off, 32);
  }
  if (lr == 0) {
#pragma unroll
    for (int k = 0; k < 8; ++k) counts[rb + k + 8 * half] = (int)cnt[k];
  }
}

// ---------------------------------------------------------------------------
// Kernel 2: exclusive prefix sum of counts[M] -> row_splits[M+1].
// Single block of 256 threads, ceil(M/256) counts per thread, LDS Hillis-Steele scan.
// ---------------------------------------------------------------------------
__global__ __launch_bounds__(256) void ns_scan_kernel(
    const int* __restrict__ counts, int* __restrict__ row_splits, int Mq) {
  __shared__ int s[256];
  const int t = threadIdx.x;
  const int chunk = (Mq + 255) / 256;
  const int lo = t * chunk;
  int sum = 0;
  for (int i = 0; i < chunk; ++i) {
    int idx = lo + i;
    if (idx < Mq) sum += counts[idx];
  }
  s[t] = sum;
  __syncthreads();
  for (int off = 1; off < 256; off <<= 1) {
    int v = (t >= off) ? s[t - off] : 0;
    __syncthreads();
    s[t] += v;
    __syncthreads();
  }
  int running = s[t] - sum;  // exclusive base for this chunk
  for (int i = 0; i < chunk; ++i) {
    int idx = lo + i;
    if (idx < Mq) {
      running += counts[idx];
      row_splits[idx + 1] = running;
    }
  }
  if (t == 0) row_splits[0] = 0;
}

// ---------------------------------------------------------------------------
// Kernel 3: emit CSR column indices in row-major nonzero order.
// Same WMMA sweep; per accumulator row, a wave32 ballot yields a 16-bit
// column mask per half; matching lanes write at row_splits[row] + cursor +
// popc(lower bits). Cursors are wave-uniform scalars; no atomics, so the
// ascending-tile / ascending-bit order reproduces jnp.nonzero exactly.
// ---------------------------------------------------------------------------
__global__ __launch_bounds__(32) void ns_fill_kernel(
    const float* __restrict__ data, const float* __restrict__ queries,
    const float* __restrict__ radi, const int* __restrict__ row_splits,
    int* __restrict__ nbr_index, int Mq, int Nd) {
  const int lane = threadIdx.x & 31;
  const int rb = blockIdx.x * 16;
  if (rb >= Mq) return;  // wave-uniform
  const int half = lane >> 4;
  const int lr = lane & 15;

  const int m = rb + lr;
  const float qx = queries[m * 3 + 0];
  const float qy = queries[m * 3 + 1];
  const float qz = queries[m * 3 + 2];
  const float q2 = qx * qx + qy * qy + qz * qz;
  v2f a;
  a.x = half ? qz : qx;
  a.y = half ? q2 : qy;

  float r2v[8];
  int rsLo[8], rsHi[8];
  unsigned curLo[8], curHi[8];
#pragma unroll
  for (int k = 0; k < 8; ++k) {
    float r = radi[rb + k + 8 * half];
    r2v[k] = r * r;
    rsLo[k] = row_splits[rb + k];
    rsHi[k] = row_splits[rb + k + 8];
    curLo[k] = 0u;
    curHi[k] = 0u;
  }
  const unsigned below = (1u << lr) - 1u;

  for (int cb = 0; cb < Nd; cb += 16) {
    const int j = cb + lr;
    __builtin_prefetch(&data[(j + 64) * 3], 0, 1);
    const float dx = data[j * 3 + 0];
    const float dy = data[j * 3 + 1];
    const float dz = data[j * 3 + 2];
    const float d2 = dx * dx + dy * dy + dz * dz;
    v2f b;
    b.x = half ? (-2.0f * dz) : (-2.0f * dx);
    b.y = half ? 1.0f : (-2.0f * dy);
    v8f c;
#pragma unroll
    for (int k = 0; k < 8; ++k) c[k] = d2;
    v8f acc = __builtin_amdgcn_wmma_f32_16x16x4_f32(
        false, a, false, b, (short)0, c, false, false);
#pragma unroll
    for (int k = 0; k < 8; ++k) {
      const bool matched = (acc[k] <= r2v[k]);
      const unsigned bal = ballot32(matched);
      const unsigned maskLo = bal & 0xFFFFu;
      const unsigned maskHi = bal >> 16;
      if (matched) {
        const unsigned myMask = half ? maskHi : maskLo;
        const int base = half ? (rsHi[k] + (int)curHi[k])
                              : (rsLo[k] + (int)curLo[k]);
        const int pos = base + (int)__popc(myMask & below);
        nbr_index[pos] = cb + lr;
      }
      curLo[k] += (unsigned)__popc(maskLo);
      curHi[k] += (unsigned)__popc(maskHi);
    }
  }
}

extern "C" void kernel_launch(void* const* d_in, const int* in_sizes, int n_in,
                              void* d_out, int out_size, void* d_ws, size_t ws_size,
                              hipStream_t stream) {
  const float* data = (const float*)d_in[0];     // [N,3] f32
  const float* queries = (const float*)d_in[1];  // [M,3] f32
  const float* radi = (const float*)d_in[2];     // [M]   f32
  const int Nd = in_sizes[0] / 3;
  const int Mq = in_sizes[2];

  int* out = (int*)d_out;
  int* row_splits = out + (out_size - (Mq + 1));  // tail of d_out
  int* nbr_index = out;                           // head of d_out
  int* counts = (int*)d_ws;                       // Mq ints of scratch

  const int rowTiles = (Mq + 15) / 16;  // one wave (= one workgroup) per tile
  ns_count_kernel<<<rowTiles, 32, 0, stream>>>(data, queries, radi, counts, Mq, Nd);
  ns_scan_kernel<<<1, 256, 0, stream>>>(counts, row_splits, Mq);
  ns_fill_kernel<<<rowTiles, 32, 0, stream>>>(data, queries, radi, row_splits,
                                              nbr_index, Mq, Nd);
}